// TrilaterationSolver_57836029608029
// MI455X (gfx1250) — compile-verified
//
#include <hip/hip_runtime.h>

// Batched 2D trilateration via Levenberg-Marquardt, one element per lane.
// Scalar VALU variant: A/B/C against packed-f32 versions showed the pk
// conversion *increases* issue slots (pair-marshalling around the scalar
// sqrt/max/select islands), so plain scalar FMA + raw v_sqrt_f32/v_rcp_f32
// (TRANS co-executes) is the minimum-issue-count form. Coalesced b64 loads,
// gfx1250 global_prefetch_b8 on the streaming inputs.

#define LM_MAX_ITER 20
#define LM_EPS      1e-6f
#define LM_INIT     0.01f
#define LM_DECAY    0.3f
#define LM_GROWTH   10.0f

__device__ __forceinline__ float frcp(float x)  { return __builtin_amdgcn_rcpf(x); }
__device__ __forceinline__ float fsqrt_(float x){ return __builtin_amdgcn_sqrtf(x); }

// Evaluate distances + residual at point (px,py) in origin-relative coords.
__device__ __forceinline__ void eval_point(
    float px, float py,
    float sax, float say, float sbx, float sby,
    float dda, float ddb,
    float& da, float& db, float& d0,
    float& f0, float& f1, float& res)
{
    float dax = px - sax, day = py - say;
    float dbx = px - sbx, dby = py - sby;
    da = fmaxf(fsqrt_(fmaf(dax, dax, day * day)), LM_EPS);
    db = fmaxf(fsqrt_(fmaf(dbx, dbx, dby * dby)), LM_EPS);
    d0 = fmaxf(fsqrt_(fmaf(px,  px,  py  * py )), LM_EPS);
    f0 = (da - d0) - dda;
    f1 = (db - d0) - ddb;
    res = fmaf(f0, f0, f1 * f1);
}

__global__ __launch_bounds__(256) void lm_trilat_kernel(
    const float2* __restrict__ sensor_a,
    const float2* __restrict__ sensor_b,
    const float2* __restrict__ sensor_origin,
    const float*  __restrict__ delta_d_a,
    const float*  __restrict__ delta_d_b,
    const float2* __restrict__ initial_guess,
    float2*       __restrict__ out,
    int n)
{
    int i = blockIdx.x * blockDim.x + threadIdx.x;
    if (i >= n) return;

    // gfx1250 streaming prefetch (global_prefetch_b8).
    __builtin_prefetch((const void*)(sensor_a + i),      0, 0);
    __builtin_prefetch((const void*)(initial_guess + i), 0, 0);

    float2 o  = sensor_origin[i];
    float2 a  = sensor_a[i];
    float2 bv = sensor_b[i];
    float2 gs = initial_guess[i];
    float dda = delta_d_a[i];
    float ddb = delta_d_b[i];

    // Origin-relative coordinates.
    float sax = a.x  - o.x, say = a.y  - o.y;
    float sbx = bv.x - o.x, sby = bv.y - o.y;
    float px  = gs.x - o.x, py  = gs.y - o.y;

    // State carried across iterations: point, distances, residual.
    float da, db, d0, f0, f1, res;
    eval_point(px, py, sax, say, sbx, sby, dda, ddb, da, db, d0, f0, f1, res);

#pragma unroll 1
    for (int it = 0; it < LM_MAX_ITER; ++it) {
        // Jacobian of [da-d0-dda, db-d0-ddb] wrt (x,y).
        float ida = frcp(da), idb = frcp(db), id0 = frcp(d0);
        float j00 = (px - sax) * ida - px * id0;
        float j01 = (py - say) * ida - py * id0;
        float j10 = (px - sbx) * idb - px * id0;
        float j11 = (py - sby) * idb - py * id0;

        // Normal equations: JtJ (symmetric 2x2) and gradient g = J^T f.
        float A  = fmaf(j00, j00, j10 * j10);   // JtJ[0][0]
        float Bc = fmaf(j00, j01, j10 * j11);   // JtJ[0][1] == JtJ[1][0]
        float D  = fmaf(j01, j01, j11 * j11);   // JtJ[1][1]
        float gx = fmaf(j00, f0,  j10 * f1);
        float gy = fmaf(j01, f0,  j11 * f1);

        float lam  = LM_INIT;
        float bpx = px, bpy = py;
        float bda = da, bdb = db, bd0 = d0;
        float bf0 = f0, bf1 = f1, bres = res;

#pragma unroll
        for (int t = 0; t < 3; ++t) {
            // Solve (JtJ + lam*I) delta = -g by Cramer's rule.
            float M00 = A + lam, M11 = D + lam;
            float det  = fmaf(M00, M11, -(Bc * Bc));
            float idet = frcp(det);
            float dx = fmaf(-M11, gx, Bc * gy) * idet;   // (-M11*gx + Bc*gy)/det
            float dy = fmaf(-M00, gy, Bc * gx) * idet;   // ( Bc*gx - M00*gy)/det

            float tx = px + dx, ty = py + dy;
            float tda, tdb, td0, tf0, tf1, tres;
            eval_point(tx, ty, sax, say, sbx, sby, dda, ddb,
                       tda, tdb, td0, tf0, tf1, tres);

            bool improved = tres < bres;   // NaN (singular det) -> not improved
            if (improved) {
                bpx = tx;  bpy = ty;
                bda = tda; bdb = tdb; bd0 = td0;
                bf0 = tf0; bf1 = tf1; bres = tres;
            }
            lam *= improved ? LM_DECAY : LM_GROWTH;
        }

        // Accept best trial (delta = 0 if nothing improved).
        px = bpx; py = bpy;
        da = bda; db = bdb; d0 = bd0;
        f0 = bf0; f1 = bf1; res = bres;
    }

    out[i] = make_float2(px + o.x, py + o.y);
}

extern "C" void kernel_launch(void* const* d_in, const int* in_sizes, int n_in,
                              void* d_out, int out_size, void* d_ws, size_t ws_size,
                              hipStream_t stream) {
    const float2* sensor_a      = (const float2*)d_in[0];
    const float2* sensor_b      = (const float2*)d_in[1];
    const float2* sensor_origin = (const float2*)d_in[2];
    const float*  delta_d_a     = (const float*) d_in[3];
    const float*  delta_d_b     = (const float*) d_in[4];
    const float2* initial_guess = (const float2*)d_in[5];
    float2*       out           = (float2*)d_out;

    int n = in_sizes[3];            // B (delta_d_a is length-B)
    const int threads = 256;        // 8 wave32 per block
    int blocks = (n + threads - 1) / threads;

    lm_trilat_kernel<<<blocks, threads, 0, stream>>>(
        sensor_a, sensor_b, sensor_origin, delta_d_a, delta_d_b,
        initial_guess, out, n);
}